// RGCNAggregator_35948876268067
// MI455X (gfx1250) — compile-verified
//
#include <hip/hip_runtime.h>

typedef __attribute__((ext_vector_type(2))) float v2f;
typedef __attribute__((ext_vector_type(8))) float v8f;

#define B_TOT 4096
#define NNB   64
#define DIM   256
#define NREL  16
#define BT    16          // batch rows per block (WMMA M)
#define PITCH 260         // DIM + 4 floats: LDS row pitch, stride%64==4 -> conflict-free b64 reads

// LDS: per-relation accumulator [NREL][BT][PITCH] + node tile [BT][PITCH]
#define PR_FLOATS   (NREL * BT * PITCH)
#define NODE_FLOATS (BT * PITCH)
#define SMEM_FLOATS (PR_FLOATS + NODE_FLOATS)

__global__ __launch_bounds__(256) void rgcn_fused_kernel(
    const float* __restrict__ node_emb,      // [1,B,D]
    const int*   __restrict__ adj_entity,    // [B,N]
    const int*   __restrict__ adj_relation,  // [B,N]
    const float* __restrict__ edge_weights,  // [B,N]
    const float* __restrict__ entity_table,  // [E,D]
    const float* __restrict__ W,             // [R,D,D]
    const float* __restrict__ res_w,         // [D,D]
    const float* __restrict__ res_b,         // [D]
    float*       __restrict__ out)           // [1,B,D]
{
    extern __shared__ float smem[];
    float* pr    = smem;              // [NREL][BT][PITCH]
    float* nodet = smem + PR_FLOATS;  // [BT][PITCH]

    const int tid   = threadIdx.x;
    const int lane  = tid & 31;
    const int wv    = tid >> 5;       // 8 waves
    const int btile = blockIdx.x;     // 256 blocks

    // ---- init LDS ----
    for (int i = tid; i < PR_FLOATS; i += 256) pr[i] = 0.0f;
    for (int i = tid; i < BT * DIM; i += 256) {
        int r = i >> 8, c = i & 255;
        nodet[r * PITCH + c] = node_emb[(size_t)(btile * BT) * DIM + i];
    }
    __syncthreads();

    // ---- Stage A: gather + weight + per-relation scatter-accumulate ----
    // wave wv owns rows 2*wv and 2*wv+1 -> exclusive LDS ownership, no atomics
    for (int rr = 0; rr < 2; ++rr) {
        const int row = wv * 2 + rr;
        const int gb  = btile * BT + row;
        for (int n = 0; n < NNB; ++n) {
            const int   ent = adj_entity[gb * NNB + n];     // wave-uniform broadcast
            const int   rel = adj_relation[gb * NNB + n];
            const float wgt = edge_weights[gb * NNB + n];
            const float* src = entity_table + (size_t)ent * DIM;
            float* dst = pr + (rel * BT + row) * PITCH;
#pragma unroll
            for (int j = 0; j < 2; ++j) {
                const int d = lane * 4 + j * 128;           // coalesced 512B per chunk
                float4 v = *(const float4*)(src + d);
                float4 c = *(const float4*)(dst + d);       // ds_load_b128
                c.x = fmaf(wgt, v.x, c.x);
                c.y = fmaf(wgt, v.y, c.y);
                c.z = fmaf(wgt, v.z, c.z);
                c.w = fmaf(wgt, v.w, c.w);
                *(float4*)(dst + d) = c;                    // ds_store_b128
            }
        }
    }
    __syncthreads();

    // ---- Stage B: fp32 WMMA grouped GEMM + residual GEMM ----
    // A 16x4 f32 frag: lane -> M = lane&15, K base = 2*(lane>>4)
    // B 4x16 f32 frag: lane -> N = lane&15, K base = 2*(lane>>4)
    const int m    = lane & 15;
    const int koff = (lane >> 4) << 1;

    for (int t = 0; t < 2; ++t) {
        const int ntile = wv + t * 8;          // 16 column tiles over 8 waves
        const int col   = ntile * 16 + m;      // output column (N) for this lane

        v8f c0 = {0.f,0.f,0.f,0.f,0.f,0.f,0.f,0.f};
        v8f c1 = {0.f,0.f,0.f,0.f,0.f,0.f,0.f,0.f};

        // residual: node_emb @ res_w^T  ->  B[k][n] = res_w[col][k] (contiguous v2f)
        {
            const float* nA = nodet + m * PITCH;
            const float* rw = res_w + (size_t)col * DIM;
            for (int k0 = 0; k0 < DIM; k0 += 8) {
                v2f a0 = *(const v2f*)(nA + k0 + koff);
                v2f b0 = *(const v2f*)(rw + k0 + koff);
                c0 = __builtin_amdgcn_wmma_f32_16x16x4_f32(
                        false, a0, false, b0, (short)0, c0, false, false);
                v2f a1 = *(const v2f*)(nA + k0 + 4 + koff);
                v2f b1 = *(const v2f*)(rw + k0 + 4 + koff);
                c1 = __builtin_amdgcn_wmma_f32_16x16x4_f32(
                        false, a1, false, b1, (short)0, c1, false, false);
            }
        }

        // messages: sum_r per_rel[r](16xD) @ W[r](DxD)
        for (int r = 0; r < NREL; ++r) {
            const float* A  = pr + (r * BT + m) * PITCH;
            const float* Wr = W + (size_t)r * DIM * DIM + col;
            for (int k0 = 0; k0 < DIM; k0 += 8) {
                const int ka = k0 + koff;
                v2f a0 = *(const v2f*)(A + ka);            // ds_load_b64, conflict-free
                v2f b0; b0.x = Wr[(size_t)ka * DIM];
                        b0.y = Wr[(size_t)(ka + 1) * DIM];
                c0 = __builtin_amdgcn_wmma_f32_16x16x4_f32(
                        false, a0, false, b0, (short)0, c0, false, false);
                const int kb = k0 + 4 + koff;
                v2f a1 = *(const v2f*)(A + kb);
                v2f b1; b1.x = Wr[(size_t)kb * DIM];
                        b1.y = Wr[(size_t)(kb + 1) * DIM];
                c1 = __builtin_amdgcn_wmma_f32_16x16x4_f32(
                        false, a1, false, b1, (short)0, c1, false, false);
            }
        }

        // ---- epilogue: relu(node + msgs + res + res_b) ----
        const float rb = res_b[col];
#pragma unroll
        for (int v = 0; v < 8; ++v) {
            const int mr = v + ((lane >> 4) << 3);         // C/D layout: M = v (+8 for hi lanes)
            float val = nodet[mr * PITCH + col] + c0[v] + c1[v] + rb;
            val = fmaxf(val, 0.0f);
            out[(size_t)(btile * BT + mr) * DIM + col] = val;
        }
    }
}

extern "C" void kernel_launch(void* const* d_in, const int* in_sizes, int n_in,
                              void* d_out, int out_size, void* d_ws, size_t ws_size,
                              hipStream_t stream) {
    const float* node_emb     = (const float*)d_in[0];
    const int*   adj_entity   = (const int*)d_in[1];
    const int*   adj_relation = (const int*)d_in[2];
    const float* edge_weights = (const float*)d_in[3];
    const float* entity_table = (const float*)d_in[4];
    const float* W            = (const float*)d_in[5];
    const float* res_w        = (const float*)d_in[6];
    const float* res_b        = (const float*)d_in[7];
    float* out = (float*)d_out;

    const size_t smem_bytes = SMEM_FLOATS * sizeof(float);   // ~276 KB of the 320 KB WGP LDS
    rgcn_fused_kernel<<<B_TOT / BT, 256, smem_bytes, stream>>>(
        node_emb, adj_entity, adj_relation, edge_weights,
        entity_table, W, res_w, res_b, out);
}